// GCN_91345364451527
// MI455X (gfx1250) — compile-verified
//
#include <hip/hip_runtime.h>

typedef __attribute__((ext_vector_type(2))) float v2f;
typedef __attribute__((ext_vector_type(8))) float v8f;

#define BN_EPS 1e-5f

// ============================ degree / CSR build ============================
__global__ void k_init_cnt(int* __restrict__ cnt, int N) {
  int i = blockIdx.x * blockDim.x + threadIdx.x;
  if (i < N) cnt[i] = 0;
}

__global__ void k_count(const int* __restrict__ dst, int* __restrict__ cnt, int E) {
  int e = blockIdx.x * blockDim.x + threadIdx.x;
  if (e < E) atomicAdd(&cnt[dst[e]], 1);   // int atomics: result order-insensitive
}

__global__ void k_dinv(const int* __restrict__ cnt, float* __restrict__ dinv, int N) {
  int i = blockIdx.x * blockDim.x + threadIdx.x;
  if (i < N) dinv[i] = rsqrtf((float)cnt[i] + 1.0f);   // +1 self-loop
}

// --- exclusive scan of cnt[N] -> rowPtr[N+1] (3 kernels, 256-wide blocks) ---
__global__ void k_scan1(const int* __restrict__ cnt, int* __restrict__ excl,
                        int* __restrict__ bsums, int N) {
  __shared__ int tmp[256];
  const int t = threadIdx.x;
  const int i = blockIdx.x * 256 + t;
  const int v = (i < N) ? cnt[i] : 0;
  tmp[t] = v;
  __syncthreads();
  for (int off = 1; off < 256; off <<= 1) {
    int add = (t >= off) ? tmp[t - off] : 0;
    __syncthreads();
    tmp[t] += add;
    __syncthreads();
  }
  if (i < N) excl[i] = tmp[t] - v;          // exclusive within block
  if (t == 255) bsums[blockIdx.x] = tmp[255];
}

__global__ void k_scan2(int* __restrict__ bsums, int nb) {   // single block, nb <= 512
  __shared__ int tmp[512];
  const int t = threadIdx.x;
  const int v = (t < nb) ? bsums[t] : 0;
  tmp[t] = v;
  __syncthreads();
  for (int off = 1; off < 512; off <<= 1) {
    int add = (t >= off) ? tmp[t - off] : 0;
    __syncthreads();
    tmp[t] += add;
    __syncthreads();
  }
  if (t < nb) bsums[t] = tmp[t] - v;        // exclusive block offsets
}

__global__ void k_scan3(const int* __restrict__ excl, const int* __restrict__ bsums,
                        int* __restrict__ rowPtr, int* __restrict__ cursor, int N, int E) {
  int i = blockIdx.x * blockDim.x + threadIdx.x;
  if (i < N) { rowPtr[i] = excl[i] + bsums[i >> 8]; cursor[i] = 0; }
  if (i == 0) rowPtr[N] = E;
}

__global__ void k_fill(const int* __restrict__ dst, const int* __restrict__ rowPtr,
                       int* __restrict__ cursor, int* __restrict__ csr, int E) {
  int e = blockIdx.x * blockDim.x + threadIdx.x;
  if (e >= E) return;
  const int d = dst[e];
  const int pos = rowPtr[d] + atomicAdd(&cursor[d], 1);
  csr[pos] = e;                              // store edge id (bucket order nondet)
}

// Sort each bucket by edge id (fixed, deterministic order), then replace id -> src node.
__global__ void k_sort_convert(const int* __restrict__ rowPtr, int* __restrict__ csr,
                               const int* __restrict__ src, int N) {
  int i = blockIdx.x * blockDim.x + threadIdx.x;
  if (i >= N) return;
  const int beg = rowPtr[i], end = rowPtr[i + 1];
  for (int a = beg + 1; a < end; ++a) {      // insertion sort, avg degree 16
    int key = csr[a];
    int b = a - 1;
    while (b >= beg && csr[b] > key) { csr[b + 1] = csr[b]; --b; }
    csr[b + 1] = key;
  }
  for (int a = beg; a < end; ++a) csr[a] = src[csr[a]];
}

// ============= fp32 WMMA GEMM: C[N x M] = A[N x K] * W[K x M] =============
__global__ __launch_bounds__(256) void k_gemm_wmma_f32(
    const float* __restrict__ A, const float* __restrict__ W, float* __restrict__ C,
    int Nrows, int K, int M, int colTiles) {
  const int lane = threadIdx.x & 31;
  const int wave = threadIdx.x >> 5;
  const int wavesPerBlock = blockDim.x >> 5;
  const int colTile = blockIdx.y * wavesPerBlock + wave;
  if (colTile >= colTiles) return;           // whole-wave exit keeps EXEC all-1 for WMMA
  const int rowTile = blockIdx.x;
  const int half = lane >> 4;
  const int l15 = lane & 15;

  int row = rowTile * 16 + l15;
  if (row >= Nrows) row = Nrows - 1;
  const int col = colTile * 16 + l15;
  const int colc = (col < M) ? col : 0;

  const float* __restrict__ Arow = A + (size_t)row * K;

  v8f acc = {0.f, 0.f, 0.f, 0.f, 0.f, 0.f, 0.f, 0.f};
  for (int k0 = 0; k0 < K; k0 += 4) {
    const int ka = k0 + (half << 1);
    v2f a = *(const v2f*)(Arow + ka);
    v2f b;
    b.x = W[(size_t)(ka + 0) * M + colc];
    b.y = W[(size_t)(ka + 1) * M + colc];
    acc = __builtin_amdgcn_wmma_f32_16x16x4_f32(
        false, a, false, b, (short)0, acc, false, false);
  }

#pragma unroll
  for (int r = 0; r < 8; ++r) {
    const int orow = rowTile * 16 + half * 8 + r;
    if (orow < Nrows && col < M)
      C[(size_t)orow * M + col] = acc[r];
  }
}

// ======================= fused gather + BN + ReLU =========================
__device__ __forceinline__ float4 f4_fma(float4 a, float c, float4 acc) {
  acc.x += a.x * c; acc.y += a.y * c; acc.z += a.z * c; acc.w += a.w * c;
  return acc;
}
__device__ __forceinline__ float4 f4_bn_relu(float4 v, float4 m, float4 vr,
                                             float4 g, float4 b) {
  float4 r;
  r.x = fmaxf((v.x - m.x) * rsqrtf(vr.x + BN_EPS) * g.x + b.x, 0.f);
  r.y = fmaxf((v.y - m.y) * rsqrtf(vr.y + BN_EPS) * g.y + b.y, 0.f);
  r.z = fmaxf((v.z - m.z) * rsqrtf(vr.z + BN_EPS) * g.z + b.z, 0.f);
  r.w = fmaxf((v.w - m.w) * rsqrtf(vr.w + BN_EPS) * g.w + b.w, 0.f);
  return r;
}

// One wave per node, 256 channels (2 x float4 per lane). No atomics; CSR order fixed.
// out = act( sum_{e in N(i)} h[src]*dinv[src]*dinv[i] + h[i]*dinv[i]^2 + bias )
__global__ __launch_bounds__(256) void k_gather256(
    const float* __restrict__ h, float* __restrict__ out,
    const int* __restrict__ rowPtr, const int* __restrict__ csrSrc,
    const float* __restrict__ dinv, const float* __restrict__ bias,
    const float* __restrict__ g, const float* __restrict__ bt,
    const float* __restrict__ mean, const float* __restrict__ var,
    int N, int doBN) {
  const int node = (int)(((size_t)blockIdx.x * blockDim.x + threadIdx.x) >> 5);
  const int lane = threadIdx.x & 31;
  if (node >= N) return;

  const float di = dinv[node];
  const float4* __restrict__ hv = (const float4*)h;
  const size_t rowBase = (size_t)node * 64;          // 256 floats = 64 float4
  const float4* __restrict__ bias4 = (const float4*)bias;

  const float selfc = di * di;
  float4 acc0 = f4_fma(hv[rowBase + lane],      selfc, bias4[lane]);
  float4 acc1 = f4_fma(hv[rowBase + lane + 32], selfc, bias4[lane + 32]);

  const int beg = rowPtr[node], end = rowPtr[node + 1];
  for (int j = beg; j < end; ++j) {
    const int s = csrSrc[j];
    const float c = dinv[s] * di;
    const float4* __restrict__ sv = hv + (size_t)s * 64;
    acc0 = f4_fma(sv[lane],      c, acc0);
    acc1 = f4_fma(sv[lane + 32], c, acc1);
  }

  if (doBN) {
    const float4* g4 = (const float4*)g;
    const float4* b4 = (const float4*)bt;
    const float4* m4 = (const float4*)mean;
    const float4* v4 = (const float4*)var;
    acc0 = f4_bn_relu(acc0, m4[lane],      v4[lane],      g4[lane],      b4[lane]);
    acc1 = f4_bn_relu(acc1, m4[lane + 32], v4[lane + 32], g4[lane + 32], b4[lane + 32]);
  }

  float4* __restrict__ ov = (float4*)out;
  ov[rowBase + lane]      = acc0;
  ov[rowBase + lane + 32] = acc1;
}

// One wave per node, 20 channels (final layer), writes d_out.
__global__ __launch_bounds__(256) void k_gather20(
    const float* __restrict__ h, float* __restrict__ out,
    const int* __restrict__ rowPtr, const int* __restrict__ csrSrc,
    const float* __restrict__ dinv, const float* __restrict__ bias, int N) {
  const int node = (int)(((size_t)blockIdx.x * blockDim.x + threadIdx.x) >> 5);
  const int lane = threadIdx.x & 31;
  if (node >= N) return;

  const float di = dinv[node];
  float acc = 0.f;
  if (lane < 20) acc = h[(size_t)node * 20 + lane] * di * di + bias[lane];

  const int beg = rowPtr[node], end = rowPtr[node + 1];
  for (int j = beg; j < end; ++j) {
    const int s = csrSrc[j];
    const float c = dinv[s] * di;
    if (lane < 20) acc += h[(size_t)s * 20 + lane] * c;
  }
  if (lane < 20) out[(size_t)node * 20 + lane] = acc;
}

// ================================ driver ==================================
extern "C" void kernel_launch(void* const* d_in, const int* in_sizes, int n_in,
                              void* d_out, int out_size, void* d_ws, size_t ws_size,
                              hipStream_t stream) {
  const float* x     = (const float*)d_in[0];
  const int*   ei    = (const int*)d_in[1];
  const float* W0    = (const float*)d_in[2];
  const float* b0    = (const float*)d_in[3];
  const float* W1    = (const float*)d_in[4];
  const float* b1    = (const float*)d_in[5];
  const float* W2    = (const float*)d_in[6];
  const float* b2    = (const float*)d_in[7];
  const float* W3    = (const float*)d_in[8];
  const float* b3    = (const float*)d_in[9];
  const float* gamma = (const float*)d_in[10];
  const float* beta  = (const float*)d_in[11];
  const float* rmean = (const float*)d_in[12];
  const float* rvar  = (const float*)d_in[13];

  const int N = in_sizes[0] / 128;   // 100000
  const int E = in_sizes[1] / 2;     // 1600000
  const int* src = ei;
  const int* dst = ei + E;

  // ---- workspace layout ----
  char* base = (char*)d_ws;
  auto alignUp = [](size_t v) { return (v + 255) & ~(size_t)255; };
  size_t off = 0;
  float* dinv   = (float*)(base + off); off = alignUp(off + (size_t)N * 4);
  int*   cnt    = (int*)(base + off);   off = alignUp(off + (size_t)N * 4);  // also cursor
  int*   excl   = (int*)(base + off);   off = alignUp(off + (size_t)N * 4);
  int*   bsums  = (int*)(base + off);   off = alignUp(off + 512 * 4);
  int*   rowPtr = (int*)(base + off);   off = alignUp(off + ((size_t)N + 1) * 4);
  int*   csr    = (int*)(base + off);   off = alignUp(off + (size_t)E * 4);
  float* bufA   = (float*)(base + off); off += (size_t)N * 256 * 4;
  float* bufB   = (float*)(base + off);
  float* out    = (float*)d_out;

  const dim3 b256(256);
  const dim3 gN((N + 255) / 256);
  const dim3 gE((E + 255) / 256);
  const int nb = (N + 255) / 256;            // scan blocks (<=512)
  const int rowTiles = (N + 15) / 16;
  const dim3 gNode((N + 7) / 8);             // 8 waves/block, 1 node per wave

  // ---- build dinv + CSR (deterministic: buckets sorted by edge id) ----
  hipLaunchKernelGGL(k_init_cnt, gN, b256, 0, stream, cnt, N);
  hipLaunchKernelGGL(k_count, gE, b256, 0, stream, dst, cnt, E);
  hipLaunchKernelGGL(k_dinv, gN, b256, 0, stream, cnt, dinv, N);
  hipLaunchKernelGGL(k_scan1, dim3(nb), b256, 0, stream, cnt, excl, bsums, N);
  hipLaunchKernelGGL(k_scan2, dim3(1), dim3(512), 0, stream, bsums, nb);
  hipLaunchKernelGGL(k_scan3, gN, b256, 0, stream, excl, bsums, rowPtr, cnt, N, E);
  hipLaunchKernelGGL(k_fill, gE, b256, 0, stream, dst, rowPtr, cnt, csr, E);
  hipLaunchKernelGGL(k_sort_convert, gN, b256, 0, stream, rowPtr, csr, src, N);

  // ---- layer 0: x(128)@W0 -> 256 ; fused gather+bias+BN0+ReLU ----
  hipLaunchKernelGGL(k_gemm_wmma_f32, dim3(rowTiles, 2), b256, 0, stream,
                     x, W0, bufA, N, 128, 256, 16);
  hipLaunchKernelGGL(k_gather256, gNode, b256, 0, stream, bufA, bufB, rowPtr, csr,
                     dinv, b0, gamma + 0, beta + 0, rmean + 0, rvar + 0, N, 1);

  // ---- layer 1 ----
  hipLaunchKernelGGL(k_gemm_wmma_f32, dim3(rowTiles, 2), b256, 0, stream,
                     bufB, W1, bufA, N, 256, 256, 16);
  hipLaunchKernelGGL(k_gather256, gNode, b256, 0, stream, bufA, bufB, rowPtr, csr,
                     dinv, b1, gamma + 256, beta + 256, rmean + 256, rvar + 256, N, 1);

  // ---- layer 2 ----
  hipLaunchKernelGGL(k_gemm_wmma_f32, dim3(rowTiles, 2), b256, 0, stream,
                     bufB, W2, bufA, N, 256, 256, 16);
  hipLaunchKernelGGL(k_gather256, gNode, b256, 0, stream, bufA, bufB, rowPtr, csr,
                     dinv, b2, gamma + 512, beta + 512, rmean + 512, rvar + 512, N, 1);

  // ---- layer 3: 256 -> 20, gather straight into d_out ----
  hipLaunchKernelGGL(k_gemm_wmma_f32, dim3(rowTiles, 1), dim3(64), 0, stream,
                     bufB, W3, bufA, N, 256, 20, 2);
  hipLaunchKernelGGL(k_gather20, gNode, b256, 0, stream, bufA, out, rowPtr, csr,
                     dinv, b3, N);
}